// Attention_49185965473844
// MI455X (gfx1250) — compile-verified
//
#include <hip/hip_runtime.h>
#include <math.h>

// Problem constants (match reference)
#define BB 64
#define SS 8192
#define DD 32
#define GBLK 8     // partial-G blocks per batch (kernel1); each block covers 1024 rows
#define SBLK 16    // s-blocks per batch (kernel3); each block covers 512 columns

typedef __attribute__((ext_vector_type(2))) float v2f;
typedef __attribute__((ext_vector_type(8))) float v8f;

// ---------------------------------------------------------------------------
// Kernel 1: partial Gram matrices  G_partial[b][blk] = sum over 1024 rows of
//           x[s,:]^T x[s,:]   (32x32 f32), via V_WMMA_F32_16X16X4_F32.
// grid = B*GBLK blocks of 256 threads (8 waves); each wave covers 128 rows.
// ---------------------------------------------------------------------------
__global__ __launch_bounds__(256) void k_gram(const float* __restrict__ x,
                                              float* __restrict__ partial)
{
    const int b    = blockIdx.x / GBLK;
    const int blk  = blockIdx.x % GBLK;
    const int wave = threadIdx.x >> 5;
    const int lane = threadIdx.x & 31;
    const int l15  = lane & 15;
    const int lh   = lane >> 4;          // 0 or 1 (lane half)

    const float* xb = x + (size_t)b * SS * DD;
    const int row0  = blk * 1024 + wave * 128;

    v8f acc0 = {}, acc1 = {}, acc2 = {}, acc3 = {}; // tiles (qi,ki)=(0,0),(0,1),(1,0),(1,1)

    for (int s = 0; s < 128; s += 4) {
        // f32 16x4 A operand == 4x16 B operand for the same column half:
        // lane L: v0 = x[s+2*lh, col], v1 = x[s+2*lh+1, col]
        const float* p = xb + (size_t)(row0 + s + 2 * lh) * DD;
        v2f lo, hi;
        lo.x = p[l15];            lo.y = p[DD + l15];        // cols 0..15
        hi.x = p[16 + l15];       hi.y = p[DD + 16 + l15];   // cols 16..31
        acc0 = __builtin_amdgcn_wmma_f32_16x16x4_f32(false, lo, false, lo, (short)0, acc0, false, false);
        acc1 = __builtin_amdgcn_wmma_f32_16x16x4_f32(false, lo, false, hi, (short)0, acc1, false, false);
        acc2 = __builtin_amdgcn_wmma_f32_16x16x4_f32(false, hi, false, lo, (short)0, acc2, false, false);
        acc3 = __builtin_amdgcn_wmma_f32_16x16x4_f32(false, hi, false, hi, (short)0, acc3, false, false);
    }

    // Cross-wave reduction via LDS: C/D layout -> VGPR j holds row (j + 8*lh), col l15
    __shared__ float red[8][4][256];
    for (int j = 0; j < 8; ++j) {
        const int m = j + 8 * lh;
        red[wave][0][m * 16 + l15] = acc0[j];
        red[wave][1][m * 16 + l15] = acc1[j];
        red[wave][2][m * 16 + l15] = acc2[j];
        red[wave][3][m * 16 + l15] = acc3[j];
    }
    __syncthreads();

    // 256 threads x 16 elements: sum 8 waves, scatter into (q,k) layout
    for (int e = 0; e < 16; ++e) {
        const int f   = threadIdx.x * 16 + e;   // 0..4095
        const int t   = f >> 8;                 // tile
        const int idx = f & 255;
        float s = 0.f;
        for (int w = 0; w < 8; ++w) s += red[w][t][idx];
        const int qi = t >> 1, ki = t & 1;
        const int m = idx >> 4, n = idx & 15;
        const int q = qi * 16 + m, k = ki * 16 + n;
        partial[((size_t)(b * GBLK + blk)) * 1024 + q * 32 + k] = s;
    }
}

// ---------------------------------------------------------------------------
// Kernel 2: per batch: G = sum partials; R = Wq^T G Wk; scores = softmax_q(R);
//           A = scores * Wv^T  -> Amat[b] (32x32). One (32,32) block per batch.
// ---------------------------------------------------------------------------
__global__ __launch_bounds__(1024) void k_attn(const float* __restrict__ partial,
                                               const float* __restrict__ Wq,
                                               const float* __restrict__ Wk,
                                               const float* __restrict__ Wv,
                                               float* __restrict__ Amat)
{
    const int b  = blockIdx.x;
    const int tx = threadIdx.x;  // column index (k, then d)
    const int ty = threadIdx.y;  // row index (q)

    __shared__ float Gs[32][32], Ts[32][32], Wqs[32][32], Wks[32][32], Wvs[32][32];

    float g = 0.f;
    for (int p = 0; p < GBLK; ++p)
        g += partial[((size_t)(b * GBLK + p)) * 1024 + ty * 32 + tx];
    Gs[ty][tx]  = g;
    Wqs[ty][tx] = Wq[ty * 32 + tx];   // W[d_in][d_out]
    Wks[ty][tx] = Wk[ty * 32 + tx];
    Wvs[ty][tx] = Wv[ty * 32 + tx];
    __syncthreads();

    float t = 0.f;                                  // T = G * Wk
    for (int d = 0; d < 32; ++d) t += Gs[ty][d] * Wks[d][tx];
    __syncthreads();
    Ts[ty][tx] = t;
    __syncthreads();

    float r = 0.f;                                  // R = Wq^T * T
    for (int d = 0; d < 32; ++d) r += Wqs[d][ty] * Ts[d][tx];
    __syncthreads();
    Gs[ty][tx] = r;
    __syncthreads();

    // softmax over q (axis=1 of [B,Dq,Dk]) for fixed column tx
    float m = -INFINITY;
    for (int qq = 0; qq < 32; ++qq) m = fmaxf(m, Gs[qq][tx]);
    float ssum = 0.f;
    for (int qq = 0; qq < 32; ++qq) ssum += __expf(Gs[qq][tx] - m);
    const float sc = __expf(r - m) / ssum;
    __syncthreads();
    Ts[ty][tx] = sc;                                // scores
    __syncthreads();

    float a = 0.f;                                  // A[q][d] = sum_k scores[q][k]*Wv[d][k]
    for (int kk = 0; kk < 32; ++kk) a += Ts[ty][kk] * Wvs[tx][kk];
    Amat[(size_t)b * 1024 + ty * 32 + tx] = a;
}

// ---------------------------------------------------------------------------
// Kernel 3: out[b] = A[b] (32x32) * x[b]^T (32 x S)  -> [B, D, S]
// grid = B*SBLK blocks of 256 threads; each wave covers 64 s-columns (4 tiles).
// ---------------------------------------------------------------------------
__global__ __launch_bounds__(256) void k_out(const float* __restrict__ x,
                                             const float* __restrict__ Amat,
                                             float* __restrict__ out)
{
    const int b    = blockIdx.x / SBLK;
    const int sb   = blockIdx.x % SBLK;
    const int wave = threadIdx.x >> 5;
    const int lane = threadIdx.x & 31;
    const int l15  = lane & 15;
    const int lh   = lane >> 4;

    const float* xb = x   + (size_t)b * SS * DD;
    const float* Ab = Amat + (size_t)b * 1024;
    float*       ob = out + (size_t)b * DD * SS;

    // Preload A operands: a[qi][kk] per f32 16x4 A layout
    v2f a[2][8];
    for (int qi = 0; qi < 2; ++qi)
        for (int kk = 0; kk < 8; ++kk) {
            const float* p = Ab + (16 * qi + l15) * 32 + 4 * kk + 2 * lh;
            a[qi][kk].x = p[0];
            a[qi][kk].y = p[1];
        }

    const int s0base = sb * 512 + wave * 64;
    for (int t = 0; t < 4; ++t) {
        const int s0 = s0base + t * 16;
        v8f acc0 = {}, acc1 = {};
        for (int kk = 0; kk < 8; ++kk) {
            // 4x16 B operand: lane L: v0 = x[s0+l15, 4kk+2lh], v1 = next col
            const float* p = xb + (size_t)(s0 + l15) * DD + 4 * kk + 2 * lh;
            v2f bv;
            bv.x = p[0];
            bv.y = p[1];
            acc0 = __builtin_amdgcn_wmma_f32_16x16x4_f32(false, a[0][kk], false, bv, (short)0, acc0, false, false);
            acc1 = __builtin_amdgcn_wmma_f32_16x16x4_f32(false, a[1][kk], false, bv, (short)0, acc1, false, false);
        }
        // C/D layout: VGPR j -> row j + 8*lh, col l15
        for (int j = 0; j < 8; ++j) {
            const int q0 = j + 8 * lh;
            ob[(size_t)q0 * SS + s0 + l15]        = acc0[j];
            ob[(size_t)(16 + q0) * SS + s0 + l15] = acc1[j];
        }
    }
}

// ---------------------------------------------------------------------------
extern "C" void kernel_launch(void* const* d_in, const int* in_sizes, int n_in,
                              void* d_out, int out_size, void* d_ws, size_t ws_size,
                              hipStream_t stream) {
    const float* x  = (const float*)d_in[0];
    const float* Wq = (const float*)d_in[1];
    const float* Wk = (const float*)d_in[2];
    const float* Wv = (const float*)d_in[3];
    float* out = (float*)d_out;

    // workspace layout: partial G [B][GBLK][32][32], then A [B][32][32]
    float* partial = (float*)d_ws;                          // 64*8*1024 f32 = 2 MB
    float* Amat    = partial + (size_t)BB * GBLK * 1024;    // 64*1024 f32 = 256 KB

    k_gram<<<dim3(BB * GBLK), dim3(256), 0, stream>>>(x, partial);
    k_attn<<<dim3(BB), dim3(32, 32), 0, stream>>>(partial, Wq, Wk, Wv, Amat);
    k_out <<<dim3(BB * SBLK), dim3(256), 0, stream>>>(x, Amat, out);
}